// RetraceLossRecursiveFast_62474594287886
// MI455X (gfx1250) — compile-verified
//
#include <hip/hip_runtime.h>
#include <cstdint>

typedef float v2f __attribute__((ext_vector_type(2)));
typedef float v8f __attribute__((ext_vector_type(8)));

namespace {
constexpr int   T      = 8192;
constexpr int   K      = 2048;
constexpr int   N      = T - 1;             // 8191 scanned rows
constexpr int   L      = 128;               // rows per chunk
constexpr int   C      = (N + L - 1) / L;   // 64 chunks
constexpr int   KB     = 256;               // threads per block
constexpr int   VEC    = 4;                 // columns per thread (float4 path)
constexpr int   COLS   = KB * VEC;          // 1024 columns per block
constexpr int   KTILES = K / COLS;          // 2
constexpr int   NBLK   = KTILES * C;        // 128 block partials
constexpr float GAMMA  = 0.99f;
constexpr int   CK     = C * K;             // 131072
}

__device__ __forceinline__ float4 ld4(const float* p) {
  return *reinterpret_cast<const float4*>(p);
}
__device__ __forceinline__ void st4(float* p, float4 v) {
  *reinterpret_cast<float4*>(p) = v;
}

// Broadcast L per-row ol values into LDS with the CDNA5 async global->LDS path.
// GVS addressing: saddr = uniform base (SGPR pair), vaddr = per-lane byte offset.
__device__ __forceinline__ void wg_load_ol_async(float* sh, const float* olbase,
                                                 int rows, int tid) {
  if (tid < rows) {
    uint32_t lds_off = (uint32_t)(uintptr_t)(sh + tid);   // low 32b = LDS byte offset
    uint32_t voff    = (uint32_t)(tid * 4);
    asm volatile("global_load_async_to_lds_b32 %0, %1, %2"
                 :
                 : "v"(lds_off), "v"(voff), "s"(olbase)
                 : "memory");
  }
  asm volatile("s_wait_asynccnt 0x0" ::: "memory");
  __syncthreads();
}

// Pass A: per (chunk c, 4 columns): chunk decay product P and chunk-local
// weighted sum S = sum_j td[j] * localcumprod[j].  All wide b128 streaming loads.
__global__ void retrace_passA(const float* __restrict__ tq,
                              const float* __restrict__ tv,
                              const float* __restrict__ r,
                              const float* __restrict__ ol,
                              const float* __restrict__ tl,
                              float* __restrict__ P,
                              float* __restrict__ S) {
  __shared__ float shOl[L];
  const int tid  = threadIdx.x;
  const int k4   = (blockIdx.x * KB + tid) * VEC;
  const int c    = blockIdx.y;
  const int row0 = c * L;
  const int rows = min(L, N - row0);
  wg_load_ol_async(shOl, ol + row0 + 1, rows, tid);

  float4 p = make_float4(1.f, 1.f, 1.f, 1.f);
  float4 s = make_float4(0.f, 0.f, 0.f, 0.f);
#pragma unroll 2
  for (int i = 0; i < rows; ++i) {
    const int    j   = row0 + i;
    const size_t o1  = (size_t)(j + 1) * K + k4;
    const size_t o0  = (size_t)j * K + k4;
    const float4 tlv = ld4(tl + o1);
    const float4 tvv = ld4(tv + o1);
    const float4 tqv = ld4(tq + o1);
    const float4 rv  = ld4(r + o0);
    const float  olv = shOl[i];
    float iw, td;
#define COMPA(c0)                                            \
    iw = __expf(fminf(tlv.c0 - olv, 0.0f));                  \
    td = rv.c0 + GAMMA * (tvv.c0 - iw * tqv.c0);             \
    p.c0 *= GAMMA * iw;                                      \
    s.c0 = fmaf(td, p.c0, s.c0);
    COMPA(x) COMPA(y) COMPA(z) COMPA(w)
#undef COMPA
  }
  st4(P + (size_t)c * K + k4, p);
  st4(S + (size_t)c * K + k4, s);
}

// Pass B: per column k, sequential over 64 chunks:
// G[c] = prod_{c'<c} P[c']   (global decay at chunk entry)
// SS[c] = sum_{c'>=c} G[c'] * S[c']   (global suffix of td*decay at chunk start)
__global__ void retrace_passB(const float* __restrict__ P,
                              const float* __restrict__ S,
                              float* __restrict__ G,
                              float* __restrict__ SS) {
  const int k = blockIdx.x * blockDim.x + threadIdx.x;
  if (k >= K) return;
  float g = 1.0f;
  for (int c = 0; c < C; ++c) {
    G[(size_t)c * K + k] = g;
    g *= P[(size_t)c * K + k];
  }
  float ss = 0.0f;
  for (int c = C - 1; c >= 0; --c) {
    ss = fmaf(G[(size_t)c * K + k], S[(size_t)c * K + k], ss);
    SS[(size_t)c * K + k] = ss;
  }
}

// Pass C: recompute iw/td streaming (b128); retrace[j] = (SS[c] - G*prefix)
// / max(G*localcumprod, 1e-10); smooth-L1 vs q; fixed-order block tree reduce.
__global__ void retrace_passC(const float* __restrict__ q_in,
                              const float* __restrict__ tq,
                              const float* __restrict__ tv,
                              const float* __restrict__ r,
                              const float* __restrict__ ol,
                              const float* __restrict__ tl,
                              const float* __restrict__ G,
                              const float* __restrict__ SS,
                              float* __restrict__ BP) {
  __shared__ float shOl[L];
  __shared__ float red[KB];
  const int tid  = threadIdx.x;
  const int k4   = (blockIdx.x * KB + tid) * VEC;
  const int c    = blockIdx.y;
  const int row0 = c * L;
  const int rows = min(L, N - row0);
  wg_load_ol_async(shOl, ol + row0 + 1, rows, tid);

  const float4 g    = ld4(G + (size_t)c * K + k4);
  const float4 ssc  = ld4(SS + (size_t)c * K + k4);
  float4 d    = make_float4(1.f, 1.f, 1.f, 1.f);   // local cumprod
  float4 pref = make_float4(0.f, 0.f, 0.f, 0.f);   // local weighted prefix (excl row)
  float4 loss = make_float4(0.f, 0.f, 0.f, 0.f);
#pragma unroll 2
  for (int i = 0; i < rows; ++i) {
    const int    j   = row0 + i;
    const size_t o1  = (size_t)(j + 1) * K + k4;
    const size_t o0  = (size_t)j * K + k4;
    const float4 tlv = ld4(tl + o1);
    const float4 tvv = ld4(tv + o1);
    const float4 tqv = ld4(tq + o1);
    const float4 rv  = ld4(r + o0);
    const float4 qv  = ld4(q_in + o0);
    const float  olv = shOl[i];
    float iw, td, gdec, rcs, ret, dl, ad;
#define COMPC(c0)                                                  \
    iw = __expf(fminf(tlv.c0 - olv, 0.0f));                        \
    td = rv.c0 + GAMMA * (tvv.c0 - iw * tqv.c0);                   \
    d.c0 *= GAMMA * iw;                                            \
    gdec = g.c0 * d.c0;                                            \
    rcs  = ssc.c0 - g.c0 * pref.c0;                                \
    ret  = rcs / fmaxf(gdec, 1e-10f);                              \
    pref.c0 = fmaf(td, d.c0, pref.c0);                             \
    dl = qv.c0 - ret;                                              \
    ad = fabsf(dl);                                                \
    loss.c0 += (ad < 1.0f) ? 0.5f * dl * dl : (ad - 0.5f);
    COMPC(x) COMPC(y) COMPC(z) COMPC(w)
#undef COMPC
  }

  red[tid] = (loss.x + loss.y) + (loss.z + loss.w);
  __syncthreads();
  for (int stride = KB / 2; stride > 0; stride >>= 1) {
    if (tid < stride) red[tid] += red[tid + stride];
    __syncthreads();
  }
  if (tid == 0) BP[blockIdx.y * gridDim.x + blockIdx.x] = red[0];
}

// Pass D: single wave32. Fixed-order lane accumulation (4 partials per lane),
// then finish the 32-lane reduction with V_WMMA_F32_16X16X4_F32 (B = ones).
__global__ void retrace_passD(const float* __restrict__ BP, float* __restrict__ out) {
  const int lane = threadIdx.x;   // 0..31, all lanes active (EXEC all 1s for WMMA)
  float s = 0.0f;
  for (int i = lane; i < NBLK; i += 32) s += BP[i];

  // A (16x4 f32): lanes 0-15 hold (M=lane, K=0/1), lanes 16-31 hold (M=lane-16, K=2/3).
  // Lane sum in K=0/2 slot, zero in K=1/3  =>  D[m,n] = s_m + s_{m+16}.
  v2f a; a.x = s;    a.y = 0.0f;
  v2f b; b.x = 1.0f; b.y = 1.0f;
  v8f cacc = {};
  cacc = __builtin_amdgcn_wmma_f32_16x16x4_f32(false, a, false, b,
                                               (short)0, cacc, false, false);
  // Lane 0 holds D[0..7, 0]; lane 16 holds D[8..15, 0].
  float t = cacc[0] + cacc[1] + cacc[2] + cacc[3] +
            cacc[4] + cacc[5] + cacc[6] + cacc[7];
  float partner = __shfl(t, 16, 32);
  if (lane == 0) {
    const float inv = 1.0f / ((float)N * (float)K);   // 1/16775168 (exact in f32)
    out[0] = (t + partner) * inv;
  }
}

extern "C" void kernel_launch(void* const* d_in, const int* in_sizes, int n_in,
                              void* d_out, int out_size, void* d_ws, size_t ws_size,
                              hipStream_t stream) {
  const float* q  = (const float*)d_in[0];  // state_trajectory_action_values (T,K)
  const float* tq = (const float*)d_in[1];  // target_state_trajectory_action_values (T,K)
  const float* tv = (const float*)d_in[2];  // target_expected_state_values (T,K)
  const float* r  = (const float*)d_in[3];  // rewards (T,K)
  const float* ol = (const float*)d_in[4];  // original_log_trajectory_action_probs (T,)
  const float* tl = (const float*)d_in[5];  // target_log_trajectory_task_action_probs (T,K)
  float* out = (float*)d_out;

  float* w  = (float*)d_ws;
  float* P  = w;            // C*K
  float* S  = w + 1 * CK;   // C*K
  float* G  = w + 2 * CK;   // C*K
  float* SS = w + 3 * CK;   // C*K
  float* BP = w + 4 * CK;   // NBLK

  dim3 gridAC(KTILES, C);   // (2, 64), 256 threads each, 4 cols/thread
  retrace_passA<<<gridAC, KB, 0, stream>>>(tq, tv, r, ol, tl, P, S);
  retrace_passB<<<K / 256, 256, 0, stream>>>(P, S, G, SS);
  retrace_passC<<<gridAC, KB, 0, stream>>>(q, tq, tv, r, ol, tl, G, SS, BP);
  retrace_passD<<<1, 32, 0, stream>>>(BP, out);
}